// AdiHs_4346506903711
// MI455X (gfx1250) — compile-verified
//
#include <hip/hip_runtime.h>
#include <math.h>

typedef __attribute__((ext_vector_type(2))) float v2f;
typedef __attribute__((ext_vector_type(8))) float v8f;

#define MIN_NORM 1e-15f
#define MAXN     (1.0f - 4e-3f)
#define ACLIP    (1.0f - 1e-7f)
#define HEADS    4

// ---------------- helpers ----------------
__device__ __forceinline__ float wred(float v) {
#pragma unroll
  for (int o = 16; o; o >>= 1) v += __shfl_xor(v, o, 32);
  return v;
}
__device__ __forceinline__ float artanh_c(float t) {
  t = fminf(fmaxf(t, -ACLIP), ACLIP);
  return 0.5f * (log1pf(t) - log1pf(-t));
}
// monotone float <-> unsigned encoding for atomicMax on signed floats
__device__ __forceinline__ unsigned fenc(float f) {
  unsigned u = __float_as_uint(f);
  return (u & 0x80000000u) ? ~u : (u | 0x80000000u);
}
__device__ __forceinline__ float fdec(unsigned u) {
  return (u & 0x80000000u) ? __uint_as_float(u & 0x7fffffffu)
                           : __uint_as_float(~u);
}

// ---------------- fp32 WMMA GEMM:  C[M,Nout] = A[M,K] @ W[Nout,K]^T (+bias) ----------------
__global__ void gemm_wmma_f32(const float* __restrict__ A,
                              const float* __restrict__ W,
                              const float* __restrict__ bias,
                              float* __restrict__ C,
                              int M, int K, int Nout) {
  const int wpb  = blockDim.x >> 5;
  const int wave = blockIdx.x * wpb + (threadIdx.x >> 5);
  const int ntiles = Nout >> 4;
  const int mtiles = (M + 15) >> 4;
  if (wave >= mtiles * ntiles) return;          // wave-uniform exit (EXEC all-1s for WMMA)
  const int tm = wave / ntiles, tn = wave % ntiles;
  const int lane = threadIdx.x & 31;
  const int half = lane >> 4;                   // 0: lanes 0-15, 1: lanes 16-31
  const int l15  = lane & 15;

  int arow = tm * 16 + l15;
  if (arow >= M) arow = M - 1;                  // clamp (stores are guarded)
  const float* __restrict__ Ar = A + (long)arow * K;            // A frag: M = l15
  const float* __restrict__ Wr = W + (long)(tn * 16 + l15) * K; // B frag: N = l15

  v8f acc = {0.f, 0.f, 0.f, 0.f, 0.f, 0.f, 0.f, 0.f};
  for (int k = 0; k < K; k += 4) {
    const int k0 = k + half * 2;                // VGPR r holds K = k + r + 2*half
    v2f a; a.x = Ar[k0]; a.y = Ar[k0 + 1];
    v2f b; b.x = Wr[k0]; b.y = Wr[k0 + 1];
    acc = __builtin_amdgcn_wmma_f32_16x16x4_f32(false, a, false, b,
                                                (short)0, acc, false, false);
  }
  const float bv = bias ? bias[tn * 16 + l15] : 0.0f;
#pragma unroll
  for (int r = 0; r < 8; ++r) {                 // D: VGPR r -> M = r + 8*half, N = l15
    const int m = tm * 16 + r + half * 8;
    if (m < M) C[(long)m * Nout + tn * 16 + l15] = acc[r] + bv;
  }
}

// ---------------- row-wise hyperbolic kernels (one wave per row) ----------------
template <int W>
__global__ void expmap0_proj_k(float* __restrict__ H, int N) {
  const int row = blockIdx.x * (blockDim.x >> 5) + (threadIdx.x >> 5);
  if (row >= N) return;
  const int lane = threadIdx.x & 31;
  float* p = H + (long)row * W;
  float v[W / 32]; float s2 = 0.f;
#pragma unroll
  for (int i = 0; i < W / 32; ++i) { v[i] = p[lane + 32 * i]; s2 += v[i] * v[i]; }
  const float n  = fmaxf(sqrtf(wred(s2)), MIN_NORM);
  float s  = tanhf(n) / n;
  const float ny = tanhf(n);
  if (ny > MAXN) s *= MAXN / ny;
#pragma unroll
  for (int i = 0; i < W / 32; ++i) p[lane + 32 * i] = v[i] * s;
}

template <int W>
__global__ void logmap0_k(float* __restrict__ H, int N) {
  const int row = blockIdx.x * (blockDim.x >> 5) + (threadIdx.x >> 5);
  if (row >= N) return;
  const int lane = threadIdx.x & 31;
  float* p = H + (long)row * W;
  float v[W / 32]; float s2 = 0.f;
#pragma unroll
  for (int i = 0; i < W / 32; ++i) { v[i] = p[lane + 32 * i]; s2 += v[i] * v[i]; }
  const float n = fmaxf(sqrtf(wred(s2)), MIN_NORM);
  const float s = artanh_c(n) / n;
#pragma unroll
  for (int i = 0; i < W / 32; ++i) p[lane + 32 * i] = v[i] * s;
}

// hyp_b = proj(expmap0(b)); also store ||hyp_b||^2 at hb[W]. Launch <<<1,32>>>.
template <int W>
__global__ void hypb_k(const float* __restrict__ b, float* __restrict__ hb) {
  const int lane = threadIdx.x & 31;
  float v[W / 32]; float s2 = 0.f;
#pragma unroll
  for (int i = 0; i < W / 32; ++i) { v[i] = b[lane + 32 * i]; s2 += v[i] * v[i]; }
  const float n = fmaxf(sqrtf(wred(s2)), MIN_NORM);
  float s  = tanhf(n) / n;
  float ny = tanhf(n);
  if (ny > MAXN) { s *= MAXN / ny; ny = MAXN; }
#pragma unroll
  for (int i = 0; i < W / 32; ++i) hb[lane + 32 * i] = v[i] * s;
  if (lane == 0) hb[W] = ny * ny;
}

// proj(mobius_add(proj(mobius_matvec(W, x)), hyp_b)) ; in-place over MX
template <int WIN, int WOUT>
__global__ void mobius_k(const float* __restrict__ X, float* __restrict__ MX,
                         const float* __restrict__ hb, int N) {
  const int row = blockIdx.x * (blockDim.x >> 5) + (threadIdx.x >> 5);
  if (row >= N) return;
  const int lane = threadIdx.x & 31;
  const float* xr = X + (long)row * WIN;
  float sx = 0.f;
#pragma unroll
  for (int i = 0; i < WIN / 32; ++i) { const float t = xr[lane + 32 * i]; sx += t * t; }
  const float xn = fmaxf(sqrtf(wred(sx)), MIN_NORM);

  float* mr = MX + (long)row * WOUT;
  float m[WOUT / 32]; float sm = 0.f;
#pragma unroll
  for (int i = 0; i < WOUT / 32; ++i) { m[i] = mr[lane + 32 * i]; sm += m[i] * m[i]; }
  const float mxn_raw = sqrtf(wred(sm));
  const float mxn = fmaxf(mxn_raw, MIN_NORM);

  float sc = tanhf(mxn / xn * artanh_c(xn)) / mxn;
  if (mxn_raw == 0.f) sc = 0.f;                 // all-zero matvec -> zero row
  float nres = sc * mxn_raw;
  if (nres > MAXN) { sc *= MAXN / nres; nres = MAXN; }

  const float b2 = hb[WOUT];
  float res[WOUT / 32], hbv[WOUT / 32]; float abp = 0.f;
#pragma unroll
  for (int i = 0; i < WOUT / 32; ++i) {
    res[i] = sc * m[i];
    hbv[i] = hb[lane + 32 * i];
    abp += res[i] * hbv[i];
  }
  const float ab = wred(abp);
  const float a2 = nres * nres;
  const float ca  = 1.f + 2.f * ab + b2;
  const float cb  = 1.f - a2;
  const float den = fmaxf(1.f + 2.f * ab + a2 * b2, MIN_NORM);

  float o[WOUT / 32]; float so = 0.f;
#pragma unroll
  for (int i = 0; i < WOUT / 32; ++i) { o[i] = (ca * res[i] + cb * hbv[i]) / den; so += o[i] * o[i]; }
  const float no = fmaxf(sqrtf(wred(so)), MIN_NORM);
  const float ps = (no > MAXN) ? MAXN / no : 1.f;
#pragma unroll
  for (int i = 0; i < WOUT / 32; ++i) mr[lane + 32 * i] = o[i] * ps;
}

// ---------------- edge kernels (one wave per edge) ----------------
template <int W, int DLOG2>
__global__ void att_alpha_k(const float* __restrict__ HT,
                            const int* __restrict__ srcp, const int* __restrict__ dstp,
                            const float* __restrict__ ai, const float* __restrict__ aj,
                            float* __restrict__ alpha, unsigned* __restrict__ amax,
                            int Etot, int E) {
  const long e = (long)blockIdx.x * (blockDim.x >> 5) + (threadIdx.x >> 5);
  if (e >= Etot) return;
  const int lane = threadIdx.x & 31;
  int s, d;
  if (e < E) { s = srcp[e]; d = dstp[e]; } else { s = d = (int)(e - E); }
  const float* xi = HT + (long)s * W;
  const float* xj = HT + (long)d * W;
  float acc[HEADS] = {0.f, 0.f, 0.f, 0.f};
#pragma unroll
  for (int i = 0; i < W / 32; ++i) {
    const int f = lane + 32 * i;
    const int h = (i * 32) >> DLOG2;            // lane-invariant head index
    acc[h] += xi[f] * ai[f] + xj[f] * aj[f];
  }
#pragma unroll
  for (int h = 0; h < HEADS; ++h) acc[h] = wred(acc[h]);
  if (lane == 0) {
#pragma unroll
    for (int h = 0; h < HEADS; ++h) {
      float a = acc[h];
      a = (a >= 0.f) ? a : 0.2f * a;            // leaky_relu 0.2
      alpha[e * HEADS + h] = a;
      atomicMax(&amax[(long)s * HEADS + h], fenc(a));
    }
  }
}

template <int W, int DLOG2>
__global__ void att_accum_k(const float* __restrict__ HT,
                            const int* __restrict__ srcp, const int* __restrict__ dstp,
                            const float* __restrict__ alpha, const unsigned* __restrict__ amax,
                            float* __restrict__ denom, float* __restrict__ aggr,
                            int Etot, int E) {
  const long e = (long)blockIdx.x * (blockDim.x >> 5) + (threadIdx.x >> 5);
  if (e >= Etot) return;
  const int lane = threadIdx.x & 31;
  int s, d;
  if (e < E) { s = srcp[e]; d = dstp[e]; } else { s = d = (int)(e - E); }
  const float* xj = HT + (long)d * W;
  float eh[HEADS];
#pragma unroll
  for (int h = 0; h < HEADS; ++h)
    eh[h] = expf(alpha[e * HEADS + h] - fdec(amax[(long)s * HEADS + h]));
  if (lane < HEADS) atomicAdd(&denom[(long)s * HEADS + lane], eh[lane]);
#pragma unroll
  for (int i = 0; i < W / 32; ++i) {
    const int f = lane + 32 * i;
    const int h = (i * 32) >> DLOG2;
    atomicAdd(&aggr[(long)s * W + f], xj[f] * eh[h]);
  }
}

// finalize (concat): u = aggr/denom; z = proj(expmap0(leaky(logmap0(proj(expmap0(u))),0.01)))
template <int W, int DLOG2>
__global__ void att_fin_concat_k(float* __restrict__ aggr, const float* __restrict__ denom, int N) {
  const int row = blockIdx.x * (blockDim.x >> 5) + (threadIdx.x >> 5);
  if (row >= N) return;
  const int lane = threadIdx.x & 31;
  float* p = aggr + (long)row * W;
  const float* dn = denom + (long)row * HEADS;
  float u[W / 32]; float s2 = 0.f;
#pragma unroll
  for (int i = 0; i < W / 32; ++i) {
    const int f = lane + 32 * i;
    u[i] = p[f] / (dn[f >> DLOG2] + 1e-16f);
    s2 += u[i] * u[i];
  }
  const float n = fmaxf(sqrtf(wred(s2)), MIN_NORM);
  float sy = tanhf(n) / n; float ny = tanhf(n);
  if (ny > MAXN) { sy *= MAXN / ny; ny = MAXN; }
  const float st = sy * artanh_c(ny) / fmaxf(ny, MIN_NORM);   // u -> tangent
  float t[W / 32]; float s3 = 0.f;
#pragma unroll
  for (int i = 0; i < W / 32; ++i) {
    float tv = u[i] * st;
    tv = (tv >= 0.f) ? tv : 0.01f * tv;
    t[i] = tv; s3 += tv * tv;
  }
  const float n3 = fmaxf(sqrtf(wred(s3)), MIN_NORM);
  float sz = tanhf(n3) / n3; const float nz = tanhf(n3);
  if (nz > MAXN) sz *= MAXN / nz;
#pragma unroll
  for (int i = 0; i < W / 32; ++i) p[lane + 32 * i] = t[i] * sz;
}

// finalize (mean over heads, d=64) + hyp_act + final logmap0 -> out[N,64]
__global__ void att_fin_mean_k(const float* __restrict__ aggr, const float* __restrict__ denom,
                               float* __restrict__ out, int N) {
  const int row = blockIdx.x * (blockDim.x >> 5) + (threadIdx.x >> 5);
  if (row >= N) return;
  const int lane = threadIdx.x & 31;
  const float* p  = aggr  + (long)row * 256;
  const float* dn = denom + (long)row * HEADS;
  float u[2]; float s2 = 0.f;
#pragma unroll
  for (int i = 0; i < 2; ++i) {
    const int dd = lane + 32 * i;
    float s = 0.f;
#pragma unroll
    for (int h = 0; h < HEADS; ++h) s += p[h * 64 + dd] / (dn[h] + 1e-16f);
    u[i] = 0.25f * s; s2 += u[i] * u[i];
  }
  const float n = fmaxf(sqrtf(wred(s2)), MIN_NORM);
  float sy = tanhf(n) / n; float ny = tanhf(n);
  if (ny > MAXN) { sy *= MAXN / ny; ny = MAXN; }
  const float st = sy * artanh_c(ny) / fmaxf(ny, MIN_NORM);
  float t[2]; float s3 = 0.f;
#pragma unroll
  for (int i = 0; i < 2; ++i) {
    float tv = u[i] * st;
    tv = (tv >= 0.f) ? tv : 0.01f * tv;
    t[i] = tv; s3 += tv * tv;
  }
  const float n3 = fmaxf(sqrtf(wred(s3)), MIN_NORM);
  float sz = tanhf(n3) / n3; float nz = tanhf(n3);
  if (nz > MAXN) { sz *= MAXN / nz; nz = MAXN; }
  const float so = sz * artanh_c(nz) / fmaxf(nz, MIN_NORM);   // final logmap0
#pragma unroll
  for (int i = 0; i < 2; ++i) out[(long)row * 64 + lane + 32 * i] = t[i] * so;
}

__global__ void zero_k(float* __restrict__ p, long n) {
  long i = (long)blockIdx.x * blockDim.x + threadIdx.x;
  const long st = (long)gridDim.x * blockDim.x;
  for (; i < n; i += st) p[i] = 0.f;
}

// ---------------- host orchestration ----------------
extern "C" void kernel_launch(void* const* d_in, const int* in_sizes, int n_in,
                              void* d_out, int out_size, void* d_ws, size_t ws_size,
                              hipStream_t stream) {
  const float* x   = (const float*)d_in[0];
  const float* Wl  = (const float*)d_in[1];   // [64,128]
  const float* bl  = (const float*)d_in[2];
  const float* W1  = (const float*)d_in[3];   // [128,64]
  const float* b1  = (const float*)d_in[4];
  const float* ai1 = (const float*)d_in[5];
  const float* aj1 = (const float*)d_in[6];
  const float* W2  = (const float*)d_in[7];   // [256,128]
  const float* b2  = (const float*)d_in[8];
  const float* ai2 = (const float*)d_in[9];
  const float* aj2 = (const float*)d_in[10];
  const int*   ei  = (const int*)d_in[11];    // [2,E] row-major

  const int NF = 128;
  const int N  = in_sizes[0] / NF;
  const int E  = in_sizes[11] / 2;
  const int Etot = E + N;                      // + self loops
  const int* srcp = ei;
  const int* dstp = ei + E;

  long o = 0;
  float* base  = (float*)d_ws;
  float* buf0  = base + o; o += (long)N * 256;
  float* buf1  = base + o; o += (long)N * 256;
  float* alpha = base + o; o += (long)Etot * HEADS;
  unsigned* amax = (unsigned*)(base + o); o += (long)N * HEADS;
  float* denom = base + o; o += (long)N * HEADS;
  float* hb    = base + o; o += 257;
  (void)ws_size; (void)n_in; (void)out_size;

  const int BLK = 256;                          // 8 waves / block
  const int rowsGrid = (N + 7) / 8;
  const int edgeGrid = (Etot + 7) / 8;
  auto wg = [](long waves) { return (int)((waves + 7) / 8); };
  const long mtiles = (long)((N + 15) / 16);

  // ---- Euclidean linear + initHyperX ----
  gemm_wmma_f32<<<wg(mtiles * (64 / 16)), BLK, 0, stream>>>(x, Wl, bl, buf0, N, NF, 64);
  expmap0_proj_k<64><<<rowsGrid, BLK, 0, stream>>>(buf0, N);

  // ---- layer1: hyp_linear (64 -> 128) ----
  gemm_wmma_f32<<<wg(mtiles * (128 / 16)), BLK, 0, stream>>>(buf0, W1, nullptr, buf1, N, 64, 128);
  hypb_k<128><<<1, 32, 0, stream>>>(b1, hb);
  mobius_k<64, 128><<<rowsGrid, BLK, 0, stream>>>(buf0, buf1, hb, N);

  // ---- layer1: attention (heads=4, d=32, concat) ----
  logmap0_k<128><<<rowsGrid, BLK, 0, stream>>>(buf1, N);
  zero_k<<<1024, 256, 0, stream>>>(buf0, (long)N * 128);
  zero_k<<<64, 256, 0, stream>>>((float*)amax, (long)N * HEADS);
  zero_k<<<64, 256, 0, stream>>>(denom, (long)N * HEADS);
  att_alpha_k<128, 5><<<edgeGrid, BLK, 0, stream>>>(buf1, srcp, dstp, ai1, aj1, alpha, amax, Etot, E);
  att_accum_k<128, 5><<<edgeGrid, BLK, 0, stream>>>(buf1, srcp, dstp, alpha, amax, denom, buf0, Etot, E);
  att_fin_concat_k<128, 5><<<rowsGrid, BLK, 0, stream>>>(buf0, denom, N);

  // ---- layer2: hyp_linear (128 -> 256) ----
  gemm_wmma_f32<<<wg(mtiles * (256 / 16)), BLK, 0, stream>>>(buf0, W2, nullptr, buf1, N, 128, 256);
  hypb_k<256><<<1, 32, 0, stream>>>(b2, hb);
  mobius_k<128, 256><<<rowsGrid, BLK, 0, stream>>>(buf0, buf1, hb, N);

  // ---- layer2: attention (heads=4, d=64, mean) + final logmap0 ----
  logmap0_k<256><<<rowsGrid, BLK, 0, stream>>>(buf1, N);
  zero_k<<<1024, 256, 0, stream>>>(buf0, (long)N * 256);
  zero_k<<<64, 256, 0, stream>>>((float*)amax, (long)N * HEADS);
  zero_k<<<64, 256, 0, stream>>>(denom, (long)N * HEADS);
  att_alpha_k<256, 6><<<edgeGrid, BLK, 0, stream>>>(buf1, srcp, dstp, ai2, aj2, alpha, amax, Etot, E);
  att_accum_k<256, 6><<<edgeGrid, BLK, 0, stream>>>(buf1, srcp, dstp, alpha, amax, denom, buf0, Etot, E);
  att_fin_mean_k<<<rowsGrid, BLK, 0, stream>>>(buf0, denom, (float*)d_out, N);
}